// SMAttentionLayer_4561255268945
// MI455X (gfx1250) — compile-verified
//
#include <hip/hip_runtime.h>
#include <hip/hip_bf16.h>
#include <math.h>

// Problem sizes (fixed by the reference)
#define Bsz 32
#define Ssz 2048
#define Hsz 1024
#define Dsz 1024
#define NCHUNK 16
#define CHUNK (Ssz / NCHUNK)   // 128 rows per chunk
#define NEG_BIG (-1e10f)

typedef __attribute__((ext_vector_type(2))) float v2f;
typedef __attribute__((ext_vector_type(8))) float v8f;

// ---------------------------------------------------------------------------
// Kernel 1: x = hidden @ W1^T   (M=32, N=1024, K=1024), fp32 WMMA 16x16x4.
// One wave per 16x16 output tile: 2 x 64 = 128 waves.
// A layout (32-bit 16x4): lane<16 -> M=lane, K = 2*(lane>>4) + vgpr.
// B layout (32-bit 4x16): lane&15 -> N,     K = 2*(lane>>4) + vgpr.
// C/D layout: vgpr r, lane group g -> M = 8*g + r, N = lane&15.
// ---------------------------------------------------------------------------
__global__ __launch_bounds__(128) void k_gemm1(const float* __restrict__ hidden,
                                               const float* __restrict__ W1,
                                               float* __restrict__ x) {
  const int lane = threadIdx.x & 31;
  const int wid  = blockIdx.x * 4 + (threadIdx.x >> 5);  // 0..127
  const int mt   = wid >> 6;   // 0..1
  const int nt   = wid & 63;   // 0..63
  const int g    = lane >> 4;
  const int l16  = lane & 15;
  const int m    = mt * 16 + l16;
  const int n    = nt * 16 + l16;

  v8f c = {};
  for (int k0 = 0; k0 < Hsz; k0 += 4) {
    const int k = k0 + 2 * g;
    v2f a, b;
    a.x = hidden[m * Hsz + k + 0];
    a.y = hidden[m * Hsz + k + 1];
    b.x = W1[n * Hsz + k + 0];          // B[k][n] = W1^T[k][n] = W1[n][k]
    b.y = W1[n * Hsz + k + 1];
    c = __builtin_amdgcn_wmma_f32_16x16x4_f32(false, a, false, b,
                                              (short)0, c, false, false);
  }
#pragma unroll
  for (int r = 0; r < 8; ++r) {
    const int mo = mt * 16 + 8 * g + r;
    x[mo * Dsz + nt * 16 + l16] = c[r];
  }
}

// ---------------------------------------------------------------------------
// Kernel 2: per (b, chunk of 128 rows):
//   scores[b,s] = enc[b,s,:] . x[b,:]   (masked-quirk applied)
//   chunk max m_c, chunk sum l_c = sum exp(score - m_c)
//   ctx_part[b,c,d] = sum_s exp(score-m_c) * enc[b,s,d]   (rows hit L2)
// Single HBM pass over the 256 MB encoder tensor.
// ---------------------------------------------------------------------------
__global__ __launch_bounds__(256) void k_scores(const float* __restrict__ enc,
                                                const float* __restrict__ x,
                                                const int* __restrict__ src_lens,
                                                float* __restrict__ scores,
                                                float* __restrict__ mc,
                                                float* __restrict__ lc,
                                                float* __restrict__ ctx_part) {
  __shared__ float xs[Dsz];
  __shared__ float sc[CHUNK];
  __shared__ float pw[CHUNK];
  __shared__ float red[256];

  const int b   = blockIdx.x / NCHUNK;
  const int cc  = blockIdx.x % NCHUNK;
  const int s0  = cc * CHUNK;
  const int tid = threadIdx.x;
  const int lane = tid & 31;
  const int wv   = tid >> 5;
  const int len  = src_lens[b];

  // stage x[b] (4 KB) in LDS
  const float4* x4  = (const float4*)(x + (size_t)b * Dsz);
  float4* xs4 = (float4*)xs;
  xs4[tid] = x4[tid];
  __syncthreads();

  const float4* enc4 = (const float4*)enc;

  // Phase A: one wave per row (strided), dot product of 1024 floats
  for (int i = wv; i < CHUNK; i += 8) {
    const int s = s0 + i;
    const size_t rb = ((size_t)b * Ssz + s) * (Dsz / 4);
    float acc = 0.f;
#pragma unroll
    for (int k = 0; k < 8; ++k) {
      float4 e  = enc4[rb + k * 32 + lane];
      float4 xx = xs4[k * 32 + lane];
      acc += e.x * xx.x + e.y * xx.y + e.z * xx.z + e.w * xx.w;
    }
#pragma unroll
    for (int off = 16; off > 0; off >>= 1) acc += __shfl_xor(acc, off, 32);
    if (lane == 0) {
      // torch quirk: masked positions AND exact-zero scores -> -1e10
      float v = (s < len && acc != 0.0f) ? acc : NEG_BIG;
      sc[i] = v;
      scores[(size_t)b * Ssz + s] = v;
    }
  }
  __syncthreads();

  // Phase B: chunk max
  float lm = -3.4e38f;
  for (int i = tid; i < CHUNK; i += 256) lm = fmaxf(lm, sc[i]);
  red[tid] = lm;
  __syncthreads();
  for (int ofs = 128; ofs > 0; ofs >>= 1) {
    if (tid < ofs) red[tid] = fmaxf(red[tid], red[tid + ofs]);
    __syncthreads();
  }
  const float m = red[0];
  __syncthreads();

  if (tid < CHUNK) pw[tid] = expf(sc[tid] - m);
  __syncthreads();

  float ls = 0.f;
  for (int i = tid; i < CHUNK; i += 256) ls += pw[i];
  red[tid] = ls;
  __syncthreads();
  for (int ofs = 128; ofs > 0; ofs >>= 1) {
    if (tid < ofs) red[tid] += red[tid + ofs];
    __syncthreads();
  }
  if (tid == 0) {
    mc[b * NCHUNK + cc] = m;
    lc[b * NCHUNK + cc] = red[0];
  }
  __syncthreads();

  // Phase C: exp-weighted partial context (rows re-read from L2)
  float4 acc4 = {0.f, 0.f, 0.f, 0.f};
  for (int i = 0; i < CHUNK; ++i) {
    const float w = pw[i];
    const size_t rb = ((size_t)b * Ssz + s0 + i) * (Dsz / 4);
    float4 e = enc4[rb + tid];
    acc4.x += w * e.x; acc4.y += w * e.y; acc4.z += w * e.z; acc4.w += w * e.w;
  }
  float4* cp4 = (float4*)ctx_part;
  cp4[((size_t)b * NCHUNK + cc) * (Dsz / 4) + tid] = acc4;
}

// ---------------------------------------------------------------------------
// Kernel 3: per b, global softmax stats M,L; emit attn.T (S,B) and per-chunk
// combine factors f_c = exp(m_c - M)/L.
// ---------------------------------------------------------------------------
__global__ __launch_bounds__(256) void k_finalize(const float* __restrict__ scores,
                                                  const float* __restrict__ mc,
                                                  const float* __restrict__ lc,
                                                  float* __restrict__ f,
                                                  float* __restrict__ attnT) {
  const int b   = blockIdx.x;
  const int tid = threadIdx.x;

  float M = -3.4e38f;
  for (int c = 0; c < NCHUNK; ++c) M = fmaxf(M, mc[b * NCHUNK + c]);
  float L = 0.f;
  for (int c = 0; c < NCHUNK; ++c) L += lc[b * NCHUNK + c] * expf(mc[b * NCHUNK + c] - M);
  const float invL = 1.0f / L;

  if (tid < NCHUNK) f[b * NCHUNK + tid] = expf(mc[b * NCHUNK + tid] - M) * invL;

  for (int s = tid; s < Ssz; s += 256) {
    attnT[(size_t)s * Bsz + b] = expf(scores[(size_t)b * Ssz + s] - M) * invL;
  }
}

// ---------------------------------------------------------------------------
// Kernel 4: ctx[b,d] = sum_c f_c * ctx_part[b,c,d]
// ---------------------------------------------------------------------------
__global__ __launch_bounds__(256) void k_combine(const float* __restrict__ ctx_part,
                                                 const float* __restrict__ f,
                                                 float* __restrict__ ctx) {
  const int b   = blockIdx.x;
  const int tid = threadIdx.x;
  const float4* cp4 = (const float4*)ctx_part;
  float4 acc = {0.f, 0.f, 0.f, 0.f};
  for (int c = 0; c < NCHUNK; ++c) {
    const float w = f[b * NCHUNK + c];
    float4 e = cp4[((size_t)b * NCHUNK + c) * (Dsz / 4) + tid];
    acc.x += w * e.x; acc.y += w * e.y; acc.z += w * e.z; acc.w += w * e.w;
  }
  ((float4*)ctx)[(size_t)b * (Dsz / 4) + tid] = acc;
}

// ---------------------------------------------------------------------------
// Kernel 5: out = tanh(concat([ctx, hidden]) @ W2^T)  (M=32, N=1024, K=2048)
// fp32 WMMA 16x16x4, same tiling as kernel 1.
// ---------------------------------------------------------------------------
__global__ __launch_bounds__(128) void k_gemm2(const float* __restrict__ ctx,
                                               const float* __restrict__ hidden,
                                               const float* __restrict__ W2,
                                               float* __restrict__ out) {
  const int lane = threadIdx.x & 31;
  const int wid  = blockIdx.x * 4 + (threadIdx.x >> 5);
  const int mt   = wid >> 6;
  const int nt   = wid & 63;
  const int g    = lane >> 4;
  const int l16  = lane & 15;
  const int m    = mt * 16 + l16;
  const int n    = nt * 16 + l16;
  const int K    = Hsz + Dsz;  // 2048

  v8f c = {};
  for (int k0 = 0; k0 < K; k0 += 4) {
    const int k = k0 + 2 * g;
    v2f a, b;
    a.x = (k + 0 < Dsz) ? ctx[m * Dsz + k + 0] : hidden[m * Hsz + (k + 0 - Dsz)];
    a.y = (k + 1 < Dsz) ? ctx[m * Dsz + k + 1] : hidden[m * Hsz + (k + 1 - Dsz)];
    b.x = W2[n * K + k + 0];
    b.y = W2[n * K + k + 1];
    c = __builtin_amdgcn_wmma_f32_16x16x4_f32(false, a, false, b,
                                              (short)0, c, false, false);
  }
#pragma unroll
  for (int r = 0; r < 8; ++r) {
    const int mo = mt * 16 + 8 * g + r;
    out[mo * Dsz + nt * 16 + l16] = tanhf(c[r]);
  }
}

// ---------------------------------------------------------------------------
// Workspace layout (floats): x | scores | mc | lc | f | ctx_part | ctx
// ---------------------------------------------------------------------------
#define X_OFF   0
#define SC_OFF  (X_OFF + Bsz * Dsz)                 // 32768
#define MC_OFF  (SC_OFF + Bsz * Ssz)                // +65536
#define LC_OFF  (MC_OFF + Bsz * NCHUNK)             // +512
#define F_OFF   (LC_OFF + Bsz * NCHUNK)             // +512
#define CP_OFF  (F_OFF + Bsz * NCHUNK)              // +512 (stays 16B aligned)
#define CTX_OFF (CP_OFF + Bsz * NCHUNK * Dsz)       // +524288

extern "C" void kernel_launch(void* const* d_in, const int* in_sizes, int n_in,
                              void* d_out, int out_size, void* d_ws, size_t ws_size,
                              hipStream_t stream) {
  const float* hidden   = (const float*)d_in[0];
  const float* enc      = (const float*)d_in[1];
  const int*   src_lens = (const int*)d_in[2];
  const float* W1       = (const float*)d_in[3];
  const float* W2       = (const float*)d_in[4];

  float* out   = (float*)d_out;            // (B, D)
  float* attnT = out + Bsz * Dsz;          // (S, B)

  float* ws       = (float*)d_ws;
  float* x        = ws + X_OFF;
  float* scores   = ws + SC_OFF;
  float* mc       = ws + MC_OFF;
  float* lc       = ws + LC_OFF;
  float* f        = ws + F_OFF;
  float* ctx_part = ws + CP_OFF;
  float* ctx      = ws + CTX_OFF;

  k_gemm1<<<32, 128, 0, stream>>>(hidden, W1, x);
  k_scores<<<Bsz * NCHUNK, 256, 0, stream>>>(enc, x, src_lens, scores, mc, lc, ctx_part);
  k_finalize<<<Bsz, 256, 0, stream>>>(scores, mc, lc, f, attnT);
  k_combine<<<Bsz, 256, 0, stream>>>(ctx_part, f, ctx);
  k_gemm2<<<32, 128, 0, stream>>>(ctx, hidden, W2, out);
}